// sub_AdversMask_3229815407244
// MI455X (gfx1250) — compile-verified
//
#include <hip/hip_runtime.h>
#include <hip/hip_bf16.h>
#include <math.h>

typedef __attribute__((ext_vector_type(2))) float v2f;
typedef __attribute__((ext_vector_type(8))) float v8f;

#define DIM 64

// ---------------- utility ----------------
__global__ void zero_f32(float* __restrict__ p, int n) {
    int i = blockIdx.x * blockDim.x + threadIdx.x;
    if (i < n) p[i] = 0.f;
}

// ---------------- degrees & norms ----------------
__global__ void degree_kernel(const int* __restrict__ src, const int* __restrict__ dst,
                              float* __restrict__ outdeg, float* __restrict__ indeg, int E) {
    int e = blockIdx.x * blockDim.x + threadIdx.x;
    if (e >= E) return;
    atomicAdd(&outdeg[src[e]], 1.0f);
    atomicAdd(&indeg[dst[e]], 1.0f);
}

__global__ void norm_kernel(const float* __restrict__ outdeg, const float* __restrict__ indeg,
                            float* __restrict__ nsrc, float* __restrict__ ndst, int N) {
    int i = blockIdx.x * blockDim.x + threadIdx.x;
    if (i >= N) return;
    float od = outdeg[i], id = indeg[i];
    nsrc[i] = (od > 0.f) ? rsqrtf(fmaxf(od, 1.f)) : 0.f;
    ndst[i] = (id > 0.f) ? rsqrtf(fmaxf(id, 1.f)) : 0.f;
}

// ---------------- edge scatter: agg[dst] += feat[src]*nsrc[src] ----------------
// 16 threads per edge, float4 per thread (64 features)
__global__ void scatter_kernel(const float* __restrict__ feat, const float* __restrict__ nsrc,
                               const int* __restrict__ src, const int* __restrict__ dst,
                               float* __restrict__ agg, int E) {
    int t = blockIdx.x * blockDim.x + threadIdx.x;
    int e = t >> 4;
    if (e >= E) return;
    int q = t & 15;
    int s = src[e], d = dst[e];
    float ns = nsrc[s];
    const float4 v = *((const float4*)(feat + (size_t)s * DIM) + q);
    float* ap = agg + (size_t)d * DIM + q * 4;
    atomicAdd(ap + 0, v.x * ns);
    atomicAdd(ap + 1, v.y * ns);
    atomicAdd(ap + 2, v.z * ns);
    atomicAdd(ap + 3, v.w * ns);
}

// ---------------- WMMA f32 GEMM: H = relu((A * ndst[:,None]) @ W + b) ----------------
// A: [N,64], W: [64,64], per-wave 16-row tile, 4 col-tiles, K stepped by 4
// v_wmma_f32_16x16x4_f32: full f32 precision (matches f32 reference).
__global__ __launch_bounds__(128) void gcn_gemm_relu(
    const float* __restrict__ A, const float* __restrict__ ndst,
    const float* __restrict__ W, const float* __restrict__ bias,
    float* __restrict__ H, int Mtiles, int N) {
    __shared__ float sW[DIM * DIM];
    __shared__ float sB[DIM];
    for (int i = threadIdx.x; i < DIM * DIM; i += 128) sW[i] = W[i];
    if (threadIdx.x < DIM) sB[threadIdx.x] = bias[threadIdx.x];
    __syncthreads();

    int wave = threadIdx.x >> 5;       // wave32
    int lane = threadIdx.x & 31;
    int mtile = blockIdx.x * 4 + wave;
    if (mtile >= Mtiles) return;       // wave-uniform: EXEC stays all-ones for WMMA

    int m0 = mtile * 16;
    int half = lane >> 4;              // 0: K=k0..k0+1, 1: K=k0+2..k0+3
    int l16 = lane & 15;
    int arow = m0 + l16;
    int arow_c = (arow < N) ? arow : (N - 1);
    float nd = (arow < N) ? ndst[arow] : 0.f;
    const float* Arow = A + (size_t)arow_c * DIM;

    v8f acc[4] = { v8f{}, v8f{}, v8f{}, v8f{} };

    for (int k0 = 0; k0 < DIM; k0 += 4) {
        int kk = k0 + half * 2;
        v2f a;
        a.x = Arow[kk] * nd;
        a.y = Arow[kk + 1] * nd;
#pragma unroll
        for (int nt = 0; nt < 4; nt++) {
            v2f b;
            b.x = sW[kk * DIM + nt * 16 + l16];
            b.y = sW[(kk + 1) * DIM + nt * 16 + l16];
            acc[nt] = __builtin_amdgcn_wmma_f32_16x16x4_f32(
                false, a, false, b, (short)0, acc[nt], false, false);
        }
    }

#pragma unroll
    for (int nt = 0; nt < 4; nt++) {
#pragma unroll
        for (int j = 0; j < 8; j++) {
            int row = m0 + j + half * 8;   // C/D layout: VGPR j -> M=j (lanes 0-15), M=j+8 (lanes 16-31)
            int col = nt * 16 + l16;
            if (row < N) {
                float v = acc[nt][j] + sB[col];
                H[(size_t)row * DIM + col] = (v > 0.f) ? v : 0.f;
            }
        }
    }
}

// ---------------- unique/compaction ----------------
__global__ void mark_present(const int* __restrict__ sid, int* __restrict__ present, int N) {
    int i = blockIdx.x * blockDim.x + threadIdx.x;
    if (i < N) present[sid[i]] = 1;
}

// single-block exclusive scan of present[] -> compact rank (unique sorts ascending ids)
__global__ void scan_present(const int* __restrict__ present, int* __restrict__ compact, int nsub) {
    __shared__ int part[256];
    int t = threadIdx.x;
    int items = (nsub + 255) / 256;
    int base = t * items;
    int local = 0;
    for (int i = 0; i < items; i++) {
        int idx = base + i;
        if (idx < nsub) local += present[idx];
    }
    part[t] = local;
    __syncthreads();
    if (t == 0) {
        int acc = 0;
        for (int i = 0; i < 256; i++) { int v = part[i]; part[i] = acc; acc += v; }
    }
    __syncthreads();
    int acc = part[t];
    for (int i = 0; i < items; i++) {
        int idx = base + i;
        if (idx < nsub) { compact[idx] = acc; acc += present[idx]; }
    }
}

// ---------------- subgraph pooling ----------------
__global__ void pool_kernel(const float* __restrict__ h, const int* __restrict__ sid,
                            const int* __restrict__ compact, float* __restrict__ sub_emb, int N) {
    int t = blockIdx.x * blockDim.x + threadIdx.x;
    int i = t >> 6;
    if (i >= N) return;
    int f = t & 63;
    int c = compact[sid[i]];
    atomicAdd(&sub_emb[(size_t)c * DIM + f], h[(size_t)i * DIM + f]);
}

// ---------------- FC head + hard gumbel-softmax (forward = one-hot argmax) ----------------
__global__ void head_kernel(const float* __restrict__ sub_emb, const float* __restrict__ fcw,
                            const float* __restrict__ fcb, const float* __restrict__ gu,
                            float* __restrict__ ysub, int nsub) {
    int s = blockIdx.x * blockDim.x + threadIdx.x;
    if (s >= nsub) return;
    float l0 = fcb[0], l1 = fcb[1];
    const float* e = sub_emb + (size_t)s * DIM;
    for (int k = 0; k < DIM; k++) {
        float v = e[k];
        l0 += v * fcw[k * 2 + 0];
        l1 += v * fcw[k * 2 + 1];
    }
    const float eps = 1e-10f;
    float g0 = -logf(-logf(gu[s * 2 + 0] + eps) + eps);
    float g1 = -logf(-logf(gu[s * 2 + 1] + eps) + eps);
    float z0 = l0 + g0, z1 = l1 + g1;
    float h0 = (z0 >= z1) ? 1.f : 0.f;  // argmax picks first on ties, like jnp.argmax
    ysub[s * 2 + 0] = h0;
    ysub[s * 2 + 1] = 1.f - h0;
}

__global__ void out_kernel(const int* __restrict__ sid, const int* __restrict__ compact,
                           const float* __restrict__ ysub, float* __restrict__ out, int N) {
    int i = blockIdx.x * blockDim.x + threadIdx.x;
    if (i >= N) return;
    int c = compact[sid[i]];
    out[i * 2 + 0] = ysub[c * 2 + 0];
    out[i * 2 + 1] = ysub[c * 2 + 1];
}

// ---------------- launch ----------------
extern "C" void kernel_launch(void* const* d_in, const int* in_sizes, int n_in,
                              void* d_out, int out_size, void* d_ws, size_t ws_size,
                              hipStream_t stream) {
    const float* x   = (const float*)d_in[0];
    const int*  edge = (const int*)d_in[1];
    const int*  sid  = (const int*)d_in[2];
    const float* W1  = (const float*)d_in[3];
    const float* b1  = (const float*)d_in[4];
    const float* W2  = (const float*)d_in[5];
    const float* b2  = (const float*)d_in[6];
    const float* fcw = (const float*)d_in[7];
    const float* fcb = (const float*)d_in[8];
    const float* gu  = (const float*)d_in[9];
    float* out = (float*)d_out;

    const int N    = in_sizes[0] / DIM;   // 100000
    const int E    = in_sizes[1] / 2;     // 1600000
    const int NSUB = in_sizes[9] / 2;     // 5000
    const int* src = edge;
    const int* dst = edge + E;

    // workspace layout (floats)
    float* ws      = (float*)d_ws;
    float* agg     = ws;                                 // N*64
    float* h       = agg + (size_t)N * DIM;              // N*64
    float* outdeg  = h + (size_t)N * DIM;                // N
    float* indeg   = outdeg + N;                         // N
    float* nsrc    = indeg + N;                          // N
    float* ndst    = nsrc + N;                           // N
    float* sub_emb = ndst + N;                           // NSUB*64
    float* ysub    = sub_emb + (size_t)NSUB * DIM;       // NSUB*2
    int* present   = (int*)(ysub + (size_t)NSUB * 2);    // NSUB
    int* compact   = present + NSUB;                     // NSUB

    const int B = 256;
    auto cdiv = [](long a, long b) { return (int)((a + b - 1) / b); };

    zero_f32<<<cdiv(2L * N, B), B, 0, stream>>>(outdeg, 2 * N);           // outdeg+indeg
    zero_f32<<<cdiv(NSUB, B), B, 0, stream>>>((float*)present, NSUB);
    degree_kernel<<<cdiv(E, B), B, 0, stream>>>(src, dst, outdeg, indeg, E);
    norm_kernel<<<cdiv(N, B), B, 0, stream>>>(outdeg, indeg, nsrc, ndst, N);

    const int Mtiles = cdiv(N, 16);
    const int gemmBlocks = cdiv(Mtiles, 4);

    // layer 1
    zero_f32<<<cdiv((long)N * DIM, B), B, 0, stream>>>(agg, N * DIM);
    scatter_kernel<<<cdiv((long)E * 16, B), B, 0, stream>>>(x, nsrc, src, dst, agg, E);
    gcn_gemm_relu<<<gemmBlocks, 128, 0, stream>>>(agg, ndst, W1, b1, h, Mtiles, N);

    // layer 2
    zero_f32<<<cdiv((long)N * DIM, B), B, 0, stream>>>(agg, N * DIM);
    scatter_kernel<<<cdiv((long)E * 16, B), B, 0, stream>>>(h, nsrc, src, dst, agg, E);
    gcn_gemm_relu<<<gemmBlocks, 128, 0, stream>>>(agg, ndst, W2, b2, h, Mtiles, N);

    // unique -> compact ids
    mark_present<<<cdiv(N, B), B, 0, stream>>>(sid, present, N);
    scan_present<<<1, 256, 0, stream>>>(present, compact, NSUB);

    // pooling + head + broadcast
    zero_f32<<<cdiv((long)NSUB * DIM, B), B, 0, stream>>>(sub_emb, NSUB * DIM);
    pool_kernel<<<cdiv((long)N * DIM, B), B, 0, stream>>>(h, sid, compact, sub_emb, N);
    head_kernel<<<cdiv(NSUB, B), B, 0, stream>>>(sub_emb, fcw, fcb, gu, ysub, NSUB);
    out_kernel<<<cdiv(N, B), B, 0, stream>>>(sid, compact, ysub, out, N);
}